// GNN_32993938768095
// MI455X (gfx1250) — compile-verified
//
#include <hip/hip_runtime.h>
#include <hip/hip_bf16.h>

// ---------------- problem constants (from reference) ----------------
#define LAYERS 2
#define ALPHA  0.8f
#define NN 100000
#define EE 800000
#define DD 64
#define RR 32
#define QQ 400000
#define SS 200000
#define BB 1024
#define HALF_E (EE / 2)

typedef float v2f __attribute__((ext_vector_type(2)));
typedef float v8f __attribute__((ext_vector_type(8)));

#define LDA 68  // 16-row A tile row stride in LDS (padded: 64+4 -> conflict-free)

// Wave-level 16x16 output tile of  A(16xK) @ B(Kx64), columns [n0, n0+16).
// A in LDS with row stride lda; B in LDS with row stride 64.
// Uses V_WMMA_F32_16X16X4_F32 (wave32), K stepped by 4.
__device__ __forceinline__ v8f wmma_tile_16xK(const float* __restrict__ sA, int lda,
                                              const float* __restrict__ sB,
                                              int K, int n0, v8f acc)
{
    const int lane = threadIdx.x & 31;
    const int hi   = lane >> 4;      // 0: lanes 0-15, 1: lanes 16-31
    const int idx  = lane & 15;
#pragma unroll
    for (int k0 = 0; k0 < 64; k0 += 4) {
        if (k0 >= K) break;
        const int ka = k0 + 2 * hi;
        v2f a, b;
        // A 16x4 fragment: lanes 0-15 -> K=k0,k0+1 ; lanes 16-31 -> K=k0+2,k0+3
        a.x = sA[idx * lda + ka];
        a.y = sA[idx * lda + ka + 1];
        // B 4x16 fragment: symmetric layout (N striped over lanes within a VGPR)
        b.x = sB[ka       * 64 + n0 + idx];
        b.y = sB[(ka + 1) * 64 + n0 + idx];
        acc = __builtin_amdgcn_wmma_f32_16x16x4_f32(false, a, false, b,
                                                    (short)0, acc, false, false);
    }
    return acc;
}

// ---------------- degree kernels ----------------
__global__ void __launch_bounds__(256)
deg_kernel(const int* __restrict__ dst, float* __restrict__ deg)
{
    for (int i = blockIdx.x * blockDim.x + threadIdx.x; i < EE;
         i += gridDim.x * blockDim.x)
        atomicAdd(&deg[dst[i]], 1.0f);
}

__global__ void __launch_bounds__(256)
invdeg_kernel(float* __restrict__ deg)
{
    for (int i = blockIdx.x * blockDim.x + threadIdx.x; i < NN;
         i += gridDim.x * blockDim.x)
        deg[i] = 1.0f / fmaxf(deg[i], 1.0f);
}

// ---------------- qualifier kernel ----------------
// buf[qual_edge] += (1-ALPHA) * (rotate(x[qual_ent], rel[qual_rel]) @ wq)
__global__ void __launch_bounds__(128)
qual_kernel(const float* __restrict__ x, const float* __restrict__ rel,
            const int* __restrict__ qual_rel, const int* __restrict__ qual_ent,
            const int* __restrict__ qual_edge,
            const float* __restrict__ wq, float* __restrict__ buf)
{
    __shared__ float sW[64 * 64];
    __shared__ float sA[16 * LDA];
    __shared__ int   sE[16];
    const int tid  = threadIdx.x;
    const int wave = tid >> 5;
    const int lane = tid & 31;
    for (int i = tid; i < 64 * 64; i += 128) sW[i] = wq[i];

    for (int tile = blockIdx.x; tile < QQ / 16; tile += gridDim.x) {
        const int q0 = tile * 16;
        __syncthreads();
        for (int i = tid; i < 16 * 64; i += 128) {
            const int r = i >> 6, d = i & 63, dd = d & 31;
            const int q   = q0 + r;
            const int ent = qual_ent[q];
            const int rl  = qual_rel[q];
            const float h_re = x[(size_t)ent * 64 + dd];
            const float h_im = x[(size_t)ent * 64 + 32 + dd];
            const float r_re = rel[(size_t)rl * 64 + dd];
            const float r_im = rel[(size_t)rl * 64 + 32 + dd];
            sA[r * LDA + d] = (d < 32) ? (h_re * r_re - h_im * r_im)
                                       : (h_re * r_im + h_im * r_re);
        }
        if (tid < 16) sE[tid] = qual_edge[q0 + tid];
        __syncthreads();

        v8f acc = {};
        acc = wmma_tile_16xK(sA, LDA, sW, 64, wave * 16, acc);

        const int hi = lane >> 4, idx = lane & 15;
        const int n  = wave * 16 + idx;
#pragma unroll
        for (int r = 0; r < 8; ++r) {
            const int m = r + 8 * hi;
            atomicAdd(&buf[(size_t)sE[m] * 64 + n], (1.0f - ALPHA) * acc[r]);
        }
    }
}

// ---------------- edge message kernel ----------------
// msg = rotate(x[src], ALPHA*rel[edge_type] + buf[e]) @ (wi|wo)
// agg[dst] += msg * inv_deg[dst]
__global__ void __launch_bounds__(128)
edge_kernel(const float* __restrict__ x, const float* __restrict__ rel,
            const int* __restrict__ src, const int* __restrict__ dst,
            const int* __restrict__ edge_type, const float* __restrict__ buf,
            const float* __restrict__ wi, const float* __restrict__ wo,
            const float* __restrict__ inv_deg, float* __restrict__ agg)
{
    __shared__ float sW[64 * 64];
    __shared__ float sA[16 * LDA];
    __shared__ int   sD[16];
    __shared__ float sS[16];
    const int tid  = threadIdx.x;
    const int wave = tid >> 5;
    const int lane = tid & 31;
    const float* W = (blockIdx.y == 0) ? wi : wo;
    for (int i = tid; i < 64 * 64; i += 128) sW[i] = W[i];
    const int ebase = blockIdx.y * HALF_E;

    for (int tile = blockIdx.x; tile < HALF_E / 16; tile += gridDim.x) {
        const int e0 = ebase + tile * 16;
        __syncthreads();
        for (int i = tid; i < 16 * 64; i += 128) {
            const int r = i >> 6, d = i & 63, dd = d & 31;
            const int e = e0 + r;
            const int s = src[e];
            const int t = edge_type[e];
            const float re_re = ALPHA * rel[(size_t)t * 64 + dd]      + buf[(size_t)e * 64 + dd];
            const float re_im = ALPHA * rel[(size_t)t * 64 + 32 + dd] + buf[(size_t)e * 64 + 32 + dd];
            const float h_re  = x[(size_t)s * 64 + dd];
            const float h_im  = x[(size_t)s * 64 + 32 + dd];
            sA[r * LDA + d] = (d < 32) ? (h_re * re_re - h_im * re_im)
                                       : (h_re * re_im + h_im * re_re);
        }
        if (tid < 16) {
            const int dn = dst[e0 + tid];
            sD[tid] = dn;
            sS[tid] = inv_deg[dn];
        }
        __syncthreads();

        v8f acc = {};
        acc = wmma_tile_16xK(sA, LDA, sW, 64, wave * 16, acc);

        const int hi = lane >> 4, idx = lane & 15;
        const int n  = wave * 16 + idx;
#pragma unroll
        for (int r = 0; r < 8; ++r) {
            const int m = r + 8 * hi;
            atomicAdd(&agg[(size_t)sD[m] * 64 + n], acc[r] * sS[m]);
        }
    }
}

// ---------------- node (self-loop + activation) kernel ----------------
// x_out = tanh((agg + rotate(x, loop_rel) @ wl) / 3 + bias) ; optional relu
__global__ void __launch_bounds__(128)
node_kernel(const float* __restrict__ x, const float* __restrict__ loop_rel,
            const float* __restrict__ wl, const float* __restrict__ agg,
            const float* __restrict__ bias, float* __restrict__ x_out,
            int apply_relu)
{
    __shared__ float sW[64 * 64];
    __shared__ float sA[16 * LDA];
    __shared__ float sLR[64];
    __shared__ float sB[64];
    const int tid  = threadIdx.x;
    const int wave = tid >> 5;
    const int lane = tid & 31;
    for (int i = tid; i < 64 * 64; i += 128) sW[i] = wl[i];
    if (tid < 64) { sLR[tid] = loop_rel[tid]; sB[tid] = bias[tid]; }

    for (int tile = blockIdx.x; tile < NN / 16; tile += gridDim.x) {
        const int n0row = tile * 16;
        __syncthreads();
        for (int i = tid; i < 16 * 64; i += 128) {
            const int r = i >> 6, d = i & 63, dd = d & 31;
            const int node = n0row + r;
            const float h_re = x[(size_t)node * 64 + dd];
            const float h_im = x[(size_t)node * 64 + 32 + dd];
            const float r_re = sLR[dd];
            const float r_im = sLR[32 + dd];
            sA[r * LDA + d] = (d < 32) ? (h_re * r_re - h_im * r_im)
                                       : (h_re * r_im + h_im * r_re);
        }
        __syncthreads();

        v8f acc = {};
        acc = wmma_tile_16xK(sA, LDA, sW, 64, wave * 16, acc);

        const int hi = lane >> 4, idx = lane & 15;
        const int n  = wave * 16 + idx;
#pragma unroll
        for (int r = 0; r < 8; ++r) {
            const int m    = r + 8 * hi;
            const int node = n0row + m;
            float v = tanhf((acc[r] + agg[(size_t)node * 64 + n]) * (1.0f / 3.0f) + sB[n]);
            if (apply_relu) v = fmaxf(v, 0.0f);
            x_out[(size_t)node * 64 + n] = v;
        }
    }
}

// ---------------- plain row GEMM (rel @ wr) ----------------
__global__ void __launch_bounds__(128)
rowmm_kernel(const float* __restrict__ A, const float* __restrict__ W,
             float* __restrict__ out, int rows)
{
    __shared__ float sW[64 * 64];
    __shared__ float sA[16 * LDA];
    const int tid  = threadIdx.x;
    const int wave = tid >> 5;
    const int lane = tid & 31;
    for (int i = tid; i < 64 * 64; i += 128) sW[i] = W[i];

    for (int tile = blockIdx.x; tile < rows / 16; tile += gridDim.x) {
        const int r0 = tile * 16;
        __syncthreads();
        for (int i = tid; i < 16 * 64; i += 128) {
            const int r = i >> 6, d = i & 63;
            sA[r * LDA + d] = A[(size_t)(r0 + r) * 64 + d];
        }
        __syncthreads();

        v8f acc = {};
        acc = wmma_tile_16xK(sA, LDA, sW, 64, wave * 16, acc);

        const int hi = lane >> 4, idx = lane & 15;
        const int n  = wave * 16 + idx;
#pragma unroll
        for (int r = 0; r < 8; ++r) {
            const int m = r + 8 * hi;
            out[(size_t)(r0 + m) * 64 + n] = acc[r];
        }
    }
}

// ---------------- mm dueling head ----------------
// in = concat(x[src[si]], rel[edge_type[si]], x[dst[si]])  (16 x 192)
// h1 = relu(in@w1+b1); h2 = relu(h1@w2+b2); out = val + adv - mean(adv)
__global__ void __launch_bounds__(128)
mm_head_kernel(const float* __restrict__ x, const float* __restrict__ rel,
               const int* __restrict__ src, const int* __restrict__ dst,
               const int* __restrict__ edge_type, const int* __restrict__ short_idx,
               const float* __restrict__ w1, const float* __restrict__ b1,
               const float* __restrict__ w2, const float* __restrict__ b2,
               const float* __restrict__ aw, const float* __restrict__ ab,
               const float* __restrict__ vw, const float* __restrict__ vb,
               float* __restrict__ out)
{
    __shared__ float sW[64 * 64];
    __shared__ float sA[16 * LDA];
    __shared__ float sH[16 * LDA];
    const int tid  = threadIdx.x;
    const int wave = tid >> 5;
    const int lane = tid & 31;
    const int hi   = lane >> 4;
    const int idx  = lane & 15;
    const int n    = wave * 16 + idx;

    for (int tile = blockIdx.x; tile < SS / 16; tile += gridDim.x) {
        const int s0 = tile * 16;
        v8f acc = {};
        // K = 192 handled as 3 chunks of 64 (w1 is (192,64) row-major)
        for (int c = 0; c < 3; ++c) {
            __syncthreads();
            for (int i = tid; i < 64 * 64; i += 128) sW[i] = w1[c * 64 * 64 + i];
            for (int i = tid; i < 16 * 64; i += 128) {
                const int r = i >> 6, d = i & 63;
                const int si = short_idx[s0 + r];
                float v;
                if (c == 0)      v = x[(size_t)src[si] * 64 + d];
                else if (c == 1) v = rel[(size_t)edge_type[si] * 64 + d];
                else             v = x[(size_t)dst[si] * 64 + d];
                sA[r * LDA + d] = v;
            }
            __syncthreads();
            acc = wmma_tile_16xK(sA, LDA, sW, 64, wave * 16, acc);
        }
        __syncthreads();
#pragma unroll
        for (int r = 0; r < 8; ++r) {
            const int m = r + 8 * hi;
            sH[m * LDA + n] = fmaxf(acc[r] + b1[n], 0.0f);
        }
        __syncthreads();
        for (int i = tid; i < 64 * 64; i += 128) sW[i] = w2[i];
        __syncthreads();
        v8f acc2 = {};
        acc2 = wmma_tile_16xK(sH, LDA, sW, 64, wave * 16, acc2);
        __syncthreads();
#pragma unroll
        for (int r = 0; r < 8; ++r) {
            const int m = r + 8 * hi;
            sA[m * LDA + n] = fmaxf(acc2[r] + b2[n], 0.0f);
        }
        __syncthreads();
        if (tid < 16) {
            float a0 = ab[0], a1 = ab[1], a2 = ab[2], vv = vb[0];
            for (int d = 0; d < 64; ++d) {
                const float h = sA[tid * LDA + d];
                a0 += h * aw[d * 3 + 0];
                a1 += h * aw[d * 3 + 1];
                a2 += h * aw[d * 3 + 2];
                vv += h * vw[d];
            }
            const float mean = (a0 + a1 + a2) * (1.0f / 3.0f);
            const size_t o = (size_t)(s0 + tid) * 3;
            out[o + 0] = vv + a0 - mean;
            out[o + 1] = vv + a1 - mean;
            out[o + 2] = vv + a2 - mean;
        }
    }
}

// ---------------- explore dueling head (B=1024 rows, 5 advantages) ----------------
__global__ void __launch_bounds__(128)
ex_head_kernel(const float* __restrict__ x, const int* __restrict__ agent_idx,
               const float* __restrict__ w1, const float* __restrict__ b1,
               const float* __restrict__ w2, const float* __restrict__ b2,
               const float* __restrict__ aw, const float* __restrict__ ab,
               const float* __restrict__ vw, const float* __restrict__ vb,
               float* __restrict__ out)
{
    __shared__ float sW[64 * 64];
    __shared__ float sA[16 * LDA];
    __shared__ float sH[16 * LDA];
    const int tid  = threadIdx.x;
    const int wave = tid >> 5;
    const int lane = tid & 31;
    const int hi   = lane >> 4;
    const int idx  = lane & 15;
    const int n    = wave * 16 + idx;

    for (int tile = blockIdx.x; tile < BB / 16; tile += gridDim.x) {
        const int a0r = tile * 16;
        __syncthreads();
        for (int i = tid; i < 64 * 64; i += 128) sW[i] = w1[i];
        for (int i = tid; i < 16 * 64; i += 128) {
            const int r = i >> 6, d = i & 63;
            sA[r * LDA + d] = x[(size_t)agent_idx[a0r + r] * 64 + d];
        }
        __syncthreads();
        v8f acc = {};
        acc = wmma_tile_16xK(sA, LDA, sW, 64, wave * 16, acc);
        __syncthreads();
#pragma unroll
        for (int r = 0; r < 8; ++r) {
            const int m = r + 8 * hi;
            sH[m * LDA + n] = fmaxf(acc[r] + b1[n], 0.0f);
        }
        __syncthreads();
        for (int i = tid; i < 64 * 64; i += 128) sW[i] = w2[i];
        __syncthreads();
        v8f acc2 = {};
        acc2 = wmma_tile_16xK(sH, LDA, sW, 64, wave * 16, acc2);
        __syncthreads();
#pragma unroll
        for (int r = 0; r < 8; ++r) {
            const int m = r + 8 * hi;
            sA[m * LDA + n] = fmaxf(acc2[r] + b2[n], 0.0f);
        }
        __syncthreads();
        if (tid < 16) {
            float a[5], vv = vb[0];
            for (int c = 0; c < 5; ++c) a[c] = ab[c];
            for (int d = 0; d < 64; ++d) {
                const float h = sA[tid * LDA + d];
                for (int c = 0; c < 5; ++c) a[c] += h * aw[d * 5 + c];
                vv += h * vw[d];
            }
            float mean = 0.0f;
            for (int c = 0; c < 5; ++c) mean += a[c];
            mean *= (1.0f / 5.0f);
            const size_t o = (size_t)(a0r + tid) * 5;
            for (int c = 0; c < 5; ++c) out[o + c] = vv + a[c] - mean;
        }
    }
}

// ---------------- host launch ----------------
extern "C" void kernel_launch(void* const* d_in, const int* in_sizes, int n_in,
                              void* d_out, int out_size, void* d_ws, size_t ws_size,
                              hipStream_t stream)
{
    (void)in_sizes; (void)n_in; (void)out_size; (void)ws_size;
    // inputs, in setup_inputs() order
    const float* x_in      = (const float*)d_in[0];
    const float* rel_in    = (const float*)d_in[1];
    const int*   edge_idx  = (const int*)d_in[2];
    const int*   edge_type = (const int*)d_in[3];
    const int*   qual_rel  = (const int*)d_in[4];
    const int*   qual_ent  = (const int*)d_in[5];
    const int*   qual_edge = (const int*)d_in[6];
    const int*   short_idx = (const int*)d_in[7];
    const int*   agent_idx = (const int*)d_in[8];
    const float* w_in      = (const float*)d_in[9];
    const float* w_out     = (const float*)d_in[10];
    const float* w_loop    = (const float*)d_in[11];
    const float* w_q       = (const float*)d_in[12];
    const float* w_rel     = (const float*)d_in[13];
    const float* loop_rel  = (const float*)d_in[14];
    const float* bias      = (const float*)d_in[15];
    const float* mm_w1     = (const float*)d_in[16];
    const float* mm_b1     = (const float*)d_in[17];
    const float* mm_w2     = (const float*)d_in[18];
    const float* mm_b2     = (const float*)d_in[19];
    const float* mm_aw     = (const float*)d_in[20];
    const float* mm_ab     = (const float*)d_in[21];
    const float* mm_vw     = (const float*)d_in[22];
    const float* mm_vb     = (const float*)d_in[23];
    const float* ex_w1     = (const float*)d_in[24];
    const float* ex_b1     = (const float*)d_in[25];
    const float* ex_w2     = (const float*)d_in[26];
    const float* ex_b2     = (const float*)d_in[27];
    const float* ex_aw     = (const float*)d_in[28];
    const float* ex_ab     = (const float*)d_in[29];
    const float* ex_vw     = (const float*)d_in[30];
    const float* ex_vb     = (const float*)d_in[31];

    const int* src = edge_idx;          // edge_index[0]
    const int* dst = edge_idx + EE;     // edge_index[1]

    // workspace carve-up
    size_t off = 0;
    auto carve = [&](size_t bytes) {
        void* p = (char*)d_ws + off;
        off += (bytes + 255) & ~(size_t)255;
        return p;
    };
    float* buf  = (float*)carve((size_t)EE * 64 * 4);  // qual_agg / rel_e partial
    float* agg  = (float*)carve((size_t)NN * 64 * 4);
    float* deg  = (float*)carve((size_t)NN * 4);
    float* xA   = (float*)carve((size_t)NN * 64 * 4);
    float* xB   = (float*)carve((size_t)NN * 64 * 4);
    float* relA = (float*)carve((size_t)RR * 64 * 4);
    float* relB = (float*)carve((size_t)RR * 64 * 4);

    // degrees (layer-invariant)
    hipMemsetAsync(deg, 0, (size_t)NN * 4, stream);
    deg_kernel<<<2048, 256, 0, stream>>>(dst, deg);
    invdeg_kernel<<<512, 256, 0, stream>>>(deg);

    const float* xc = x_in;
    const float* rc = rel_in;
    float* xn[2]  = {xA, xB};
    float* rn[2]  = {relA, relB};

    for (int l = 0; l < LAYERS; ++l) {
        hipMemsetAsync(buf, 0, (size_t)EE * 64 * 4, stream);
        hipMemsetAsync(agg, 0, (size_t)NN * 64 * 4, stream);

        qual_kernel<<<2048, 128, 0, stream>>>(xc, rc, qual_rel, qual_ent, qual_edge,
                                              w_q + (size_t)l * 64 * 64, buf);
        edge_kernel<<<dim3(1024, 2), 128, 0, stream>>>(xc, rc, src, dst, edge_type, buf,
                                                       w_in  + (size_t)l * 64 * 64,
                                                       w_out + (size_t)l * 64 * 64,
                                                       deg, agg);
        node_kernel<<<1024, 128, 0, stream>>>(xc, loop_rel + (size_t)l * 64,
                                              w_loop + (size_t)l * 64 * 64, agg,
                                              bias + (size_t)l * 64, xn[l],
                                              (l < LAYERS - 1) ? 1 : 0);
        rowmm_kernel<<<2, 128, 0, stream>>>(rc, w_rel + (size_t)l * 64 * 64, rn[l], RR);

        xc = xn[l];
        rc = rn[l];
    }

    float* out = (float*)d_out;
    mm_head_kernel<<<2048, 128, 0, stream>>>(xc, rc, src, dst, edge_type, short_idx,
                                             mm_w1, mm_b1, mm_w2, mm_b2,
                                             mm_aw, mm_ab, mm_vw, mm_vb, out);
    ex_head_kernel<<<64, 128, 0, stream>>>(xc, agent_idx,
                                           ex_w1, ex_b1, ex_w2, ex_b2,
                                           ex_aw, ex_ab, ex_vw, ex_vb,
                                           out + (size_t)SS * 3);
}